// HungarianMatcher_2499670966616
// MI455X (gfx1250) — compile-verified
//
#include <hip/hip_runtime.h>
#include <math.h>

// Problem constants (from reference): BS=64, Q=128, NBOX=64, NCLS=64, T=32
#define BSZ  64
#define QN   128
#define NBOX 64
#define NCLS 64
#define TN   32

typedef __attribute__((ext_vector_type(2))) float v2f;
typedef __attribute__((ext_vector_type(8))) float v8f;

// Padded LDS row strides (bank-conflict avoidance: +4 floats -> 16 strided
// rows hit banks {0,4,...,60}; 68 floats = 272 B keeps 16-B chunk alignment)
#define SIMS 68   // membership [Q][NBOX] rows (raw; dummy folded into B/norms)
#define TGS  68   // tgt        [T][NBOX] rows
#define CTS  132  // Ct         [T][Q]    rows (cost transposed for the solver)

// Intra-wave LDS sync: CDNA5 split dependency counter for DS ops.
__device__ __forceinline__ void wave_lds_sync() {
  __builtin_amdgcn_wave_barrier();
  asm volatile("s_wait_dscnt 0" ::: "memory");
  __builtin_amdgcn_wave_barrier();
}

// Async global->LDS copy, 16 B per lane (GLOBAL_LOAD_ASYNC_TO_LDS_B128,
// tracked by ASYNCcnt). lds_byte = wave-relative LDS byte offset (= low 32
// bits of the flat shared-pointer per the LDS aperture mapping).
__device__ __forceinline__ void async_g2l_b128(unsigned lds_byte,
                                               const float* g) {
  asm volatile("global_load_async_to_lds_b128 %0, %1, off"
               :: "v"(lds_byte), "v"(g) : "memory");
}
__device__ __forceinline__ void wait_async0() {
  asm volatile("s_wait_asynccnt 0" ::: "memory");
}
__device__ __forceinline__ unsigned lds_off32(const void* p) {
  return (unsigned)(unsigned long long)(uintptr_t)p;
}

__global__ __launch_bounds__(256)
void hungarian_fused(const float* __restrict__ pred,       // [BS,Q,NCLS]
                     const float* __restrict__ membership, // [BS,Q,NBOX]
                     const float* __restrict__ dummy,      // [BS,NBOX]
                     const float* __restrict__ tgtm,       // [BS,T,NBOX]
                     const int*   __restrict__ tgtc,       // [BS,T]
                     int* __restrict__ out_rows,           // [BS,T]
                     int* __restrict__ out_cols)           // [BS,T]
{
  __shared__ alignas(16) float memLDS [QN * SIMS]; // raw membership
  __shared__ alignas(16) float tgtLDS [TN * TGS];  // raw tgt rows
  __shared__ alignas(16) float tgtSLDS[TN * TGS];  // tgt * dummy (WMMA B)
  __shared__ float  ctLDS [TN * CTS];              // Ct[t][q], transposed cost
  __shared__ float  snq[QN], pmax[QN], psum[QN];
  __shared__ float  snt[TN];
  __shared__ float  dumLDS[NBOX];
  __shared__ int    clsLDS[TN];
  __shared__ double uArr[TN + 1];       // row potentials (solver rows = t)
  __shared__ int    pArr[QN + 1];       // column -> assigned row (1-based)
  __shared__ int    wayArr[QN + 1];     // augmenting-path back pointers

  const int b    = blockIdx.x;
  const int tid  = threadIdx.x;
  const int wave = tid >> 5;
  const int lane = tid & 31;
  const int lhi  = lane >> 4;   // which K-half of the wave
  const int llo  = lane & 15;   // M (A) / N (B) index within fragment

  // ------------- Phase A0: kick off async HBM->LDS staging (ASYNCcnt) ------
  // membership[b]: 128 rows x 16 chunks of 16 B  -> 8 chunks/thread
  const unsigned memBase = lds_off32(&memLDS[0]);
  #pragma unroll
  for (int it = 0; it < 8; ++it) {
    const int e = tid + 256 * it;        // chunk id
    const int q = e >> 4, c = (e & 15) * 4;
    async_g2l_b128(memBase + (unsigned)(q * SIMS + c) * 4u,
                   membership + ((size_t)b * QN + q) * NBOX + c);
  }
  // tgt[b]: 32 rows x 16 chunks -> 2 chunks/thread
  const unsigned tgtBase = lds_off32(&tgtLDS[0]);
  #pragma unroll
  for (int it = 0; it < 2; ++it) {
    const int e = tid + 256 * it;
    const int t = e >> 4, c = (e & 15) * 4;
    async_g2l_b128(tgtBase + (unsigned)(t * TGS + c) * 4u,
                   tgtm + ((size_t)b * TN + t) * NBOX + c);
  }
  if (tid < NBOX) dumLDS[tid] = dummy[b * NBOX + tid];
  if (tid < TN)   clsLDS[tid] = tgtc[b * TN + tid];

  // ------------- Phase A1: softmax stats (overlaps the async DMA) ----------
  if (tid < QN) {
    const float* pr = pred + ((size_t)b * QN + tid) * NCLS;
    float m = -INFINITY;
    for (int c = 0; c < NCLS; ++c) m = fmaxf(m, pr[c]);
    float e = 0.f;
    for (int c = 0; c < NCLS; ++c) e += expf(pr[c] - m);
    pmax[tid] = m; psum[tid] = e;
  }
  wait_async0();        // each wave drains its own async copies
  __syncthreads();      // cross-wave visibility of all staged LDS

  // ------------- Phase A2: norms + dummy-scaled B operand ------------------
  if (tid < QN) {
    float s = 0.f;
    for (int n = 0; n < NBOX; ++n) {
      float x = memLDS[tid * SIMS + n] * dumLDS[n];   // sim element
      s += x * x;
    }
    snq[tid] = s;
  }
  if (tid >= QN && tid < QN + TN) {
    int t = tid - QN;                                  // raw tgt norm
    float s = 0.f;
    for (int n = 0; n < NBOX; ++n) { float x = tgtLDS[t * TGS + n]; s += x * x; }
    snt[t] = s;
  }
  for (int idx = tid; idx < TN * NBOX; idx += 256) {   // B = tgt * dummy
    int t = idx >> 6, n = idx & 63;
    tgtSLDS[t * TGS + n] = tgtLDS[t * TGS + n] * dumLDS[n];
  }
  __syncthreads();

  // ------------- Phase B: f32 WMMA Gram term + cost assembly ---------------
  // dot = mem @ (d*tgt)^T = sim @ tgt^T  (Q x T, K = NBOX), 16x16 tiles, K=4.
  for (int tile = wave; tile < 16; tile += 8) {
    const int qt = tile >> 1, tt = tile & 1;
    v8f acc = {0.f, 0.f, 0.f, 0.f, 0.f, 0.f, 0.f, 0.f};
    const float* aRow = &memLDS [(qt * 16 + llo) * SIMS]; // A: row M = llo
    const float* bRow = &tgtSLDS[(tt * 16 + llo) * TGS];  // B: col N = llo
    for (int k0 = 0; k0 < NBOX; k0 += 4) {
      const int kb = k0 + lhi * 2;
      v2f a;  a.x  = aRow[kb]; a.y  = aRow[kb + 1];   // A[m][k], A[m][k+1]
      v2f bb; bb.x = bRow[kb]; bb.y = bRow[kb + 1];   // B[k][n], B[k+1][n]
      acc = __builtin_amdgcn_wmma_f32_16x16x4_f32(
          false, a, false, bb, (short)0, acc, false, false);
    }
    const int   tcol = tt * 16 + llo;
    const int   cls  = clsLDS[tcol];
    const float sn_t = snt[tcol];
    #pragma unroll
    for (int r = 0; r < 8; ++r) {
      const int q = qt * 16 + r + 8 * lhi;           // D layout: M = r + 8*hi
      float code = sqrtf(fmaxf(snq[q] + sn_t - 2.0f * acc[r], 0.f));
      float pcls =
          expf(pred[((size_t)b * QN + q) * NCLS + cls] - pmax[q]) / psum[q];
      ctLDS[tcol * CTS + q] = code - pcls;           // COST_CLASS=COST_CODE=1
    }
  }
  __syncthreads();

  // ------------- Phase C: Jonker-Volgenant LSA on wave 0 -------------------
  // Solver works on C^T (n=T rows, m=Q cols), matching the reference's
  // transposed path. Lane l owns columns j = l+1+32s (s=0..3); lane 0 also
  // owns the virtual column j=0. Column state lives in registers.
  if (tid >= 32) return;

  double vR[4]    = {0.0, 0.0, 0.0, 0.0};
  double minvR[4];
  bool   usedR[4];
  bool   used0 = false;
  double v0    = 0.0;

  for (int j = lane; j <= QN; j += 32) pArr[j] = 0;
  for (int i = lane; i <= TN; i += 32) uArr[i] = 0.0;
  wave_lds_sync();

  for (int i = 1; i <= TN; ++i) {
    #pragma unroll
    for (int s = 0; s < 4; ++s) { minvR[s] = INFINITY; usedR[s] = false; }
    used0 = false;
    if (lane == 0) pArr[0] = i;
    wave_lds_sync();

    int j0 = 0;
    while (true) {
      // mark used[j0] (owner lane only)
      if (j0 == 0) { if (lane == 0) used0 = true; }
      else if (((j0 - 1) & 31) == lane) usedR[(j0 - 1) >> 5] = true;

      const int    i0   = pArr[j0];
      const double ui0  = uArr[i0];
      const float* crow = &ctLDS[(i0 - 1) * CTS];

      double bestv = INFINITY; int bestj = QN + 1;
      #pragma unroll
      for (int s = 0; s < 4; ++s) {
        const int j = lane + 1 + 32 * s;
        if (!usedR[s]) {
          double cur = (double)crow[j - 1] - ui0 - vR[s];
          if (cur < minvR[s]) { minvR[s] = cur; wayArr[j] = j0; }
          if (minvR[s] < bestv) { bestv = minvR[s]; bestj = j; }
        }
      }
      // wave argmin with first-index tie-break (matches np.argmin)
      for (int off = 16; off; off >>= 1) {
        double ov = __shfl_xor(bestv, off, 32);
        int    oj = __shfl_xor(bestj, off, 32);
        if (ov < bestv || (ov == bestv && oj < bestj)) { bestv = ov; bestj = oj; }
      }
      const double delta = bestv;
      const int    j1    = bestj;

      // potential updates (rows on the path are distinct -> no u collisions)
      if (lane == 0 && used0) { uArr[pArr[0]] += delta; v0 -= delta; }
      #pragma unroll
      for (int s = 0; s < 4; ++s) {
        const int j = lane + 1 + 32 * s;
        if (usedR[s]) { uArr[pArr[j]] += delta; vR[s] -= delta; }
        else          { minvR[s] -= delta; }
      }
      wave_lds_sync();

      j0 = j1;
      if (pArr[j0] == 0) break;
    }

    // augment along back pointers (serial chain, lane 0)
    if (lane == 0) {
      int jj = j0;
      while (jj) { int jp = wayArr[jj]; pArr[jj] = pArr[jp]; jj = jp; }
    }
    wave_lds_sync();
  }
  (void)v0;

  // Emit: ascending q (== ascending j) gives the reference's argsort(rows).
  if (lane == 0) {
    int k = 0;
    for (int j = 1; j <= QN; ++j) {
      const int pv = pArr[j];
      if (pv != 0) {
        out_rows[b * TN + k] = j - 1;   // q index
        out_cols[b * TN + k] = pv - 1;  // t index
        ++k;
      }
    }
  }
}

extern "C" void kernel_launch(void* const* d_in, const int* in_sizes, int n_in,
                              void* d_out, int out_size, void* d_ws, size_t ws_size,
                              hipStream_t stream) {
  const float* pred       = (const float*)d_in[0];
  const float* membership = (const float*)d_in[1];
  const float* dummy      = (const float*)d_in[2];
  const float* tgtm       = (const float*)d_in[3];
  const int*   tgtc       = (const int*)d_in[4];
  int* out = (int*)d_out;            // [rows(BS*T) | cols(BS*T)] int32
  hungarian_fused<<<dim3(BSZ), dim3(256), 0, stream>>>(
      pred, membership, dummy, tgtm, tgtc, out, out + BSZ * TN);
}